// EnhancedDeformableAttention_44212393345414
// MI455X (gfx1250) — compile-verified
//
#include <hip/hip_runtime.h>
#include <hip/hip_bf16.h>

#define B_   4
#define LQ_  2048
#define C_   256
#define NH_  8
#define NL_  4
#define NP_  8
#define HD_  32
#define LV_  21760

typedef __attribute__((ext_vector_type(2))) float v2f;
typedef __attribute__((ext_vector_type(8))) float v8f;

// -----------------------------------------------------------------------------
// Generic fp32 WMMA GEMM:  out[M,N] = scale*(A[M,K] @ W[K,N] + bias[N]) (+relu)
//                          (+ addend[M,N] if non-null)
// One wave computes a 64x32 output tile: 4 M-subtiles x 2 N-subtiles of
// V_WMMA_F32_16X16X4_F32. Per k-step: 8 WMMAs vs 4 A b64-loads + 4 B b32-loads
// (B fragments amortized over the 4 M-subtiles).
// Requires M%64==0, N%32==0, K%4==0 — true for every call site here.
// Early exits are wave-uniform -> EXEC all-ones at every WMMA (ISA req).
// -----------------------------------------------------------------------------
__global__ __launch_bounds__(256) void gemm_f32_wmma(
    const float* __restrict__ A, const float* __restrict__ W,
    const float* __restrict__ bias, const float* __restrict__ addend,
    float* __restrict__ out, int M, int N, int K, float scale, int relu)
{
    const int lane   = threadIdx.x & 31;
    const int waveId = threadIdx.x >> 5;
    const int tilesN = N >> 5;
    const int tile   = blockIdx.x * (blockDim.x >> 5) + waveId;
    if (tile >= (M >> 6) * tilesN) return;          // wave-uniform guard
    const int m0 = (tile / tilesN) << 6;            // 64-row macro tile
    const int n0 = (tile % tilesN) << 5;            // 32-col macro tile

    const int l15   = lane & 15;
    const int khalf = (lane >> 4) << 1;             // 0 for lanes 0-15, 2 for 16-31

    v8f acc00 = {}, acc01 = {};
    v8f acc10 = {}, acc11 = {};
    v8f acc20 = {}, acc21 = {};
    v8f acc30 = {}, acc31 = {};

    // A fragment (per M-subtile i): lane L(0-15) row m0+16i+L, K = k..k+1;
    // lanes 16-31 hold K = k+2..k+3 (ISA 16x4 f32 A layout).
    const float* a0 = A + (size_t)(m0 +  0 + l15) * K + khalf;
    const float* a1 = A + (size_t)(m0 + 16 + l15) * K + khalf;
    const float* a2 = A + (size_t)(m0 + 32 + l15) * K + khalf;
    const float* a3 = A + (size_t)(m0 + 48 + l15) * K + khalf;
    // B fragment: lane L(0-15) col n0+L rows K=k..k+1; lanes 16-31 K=k+2..k+3.
    const float* wp  = W + (size_t)khalf * N + (n0 + l15);
    const float* wpn = wp + N;                      // row k+1 (resp. k+3)

    for (int k = 0; k < K; k += 4) {
        const v2f fa0 = *(const v2f*)(a0 + k);      // 8B aligned: even float idx
        const v2f fa1 = *(const v2f*)(a1 + k);
        const v2f fa2 = *(const v2f*)(a2 + k);
        const v2f fa3 = *(const v2f*)(a3 + k);
        v2f b0, b1;
        b0.x = wp[0];   b0.y = wpn[0];
        b1.x = wp[16];  b1.y = wpn[16];
        wp  += (size_t)4 * N;
        wpn += (size_t)4 * N;
        acc00 = __builtin_amdgcn_wmma_f32_16x16x4_f32(false, fa0, false, b0, (short)0, acc00, false, false);
        acc01 = __builtin_amdgcn_wmma_f32_16x16x4_f32(false, fa0, false, b1, (short)0, acc01, false, false);
        acc10 = __builtin_amdgcn_wmma_f32_16x16x4_f32(false, fa1, false, b0, (short)0, acc10, false, false);
        acc11 = __builtin_amdgcn_wmma_f32_16x16x4_f32(false, fa1, false, b1, (short)0, acc11, false, false);
        acc20 = __builtin_amdgcn_wmma_f32_16x16x4_f32(false, fa2, false, b0, (short)0, acc20, false, false);
        acc21 = __builtin_amdgcn_wmma_f32_16x16x4_f32(false, fa2, false, b1, (short)0, acc21, false, false);
        acc30 = __builtin_amdgcn_wmma_f32_16x16x4_f32(false, fa3, false, b0, (short)0, acc30, false, false);
        acc31 = __builtin_amdgcn_wmma_f32_16x16x4_f32(false, fa3, false, b1, (short)0, acc31, false, false);
    }

    // C/D layout: VGPR r -> row +r (lanes 0-15) / +r+8 (lanes 16-31),
    // col = n0 + (lane&15) (+16 for the second N-subtile).
    const int rhalf = (lane >> 4) << 3;
    const int col   = n0 + l15;
    const float bi0 = bias ? bias[col]      : 0.f;
    const float bi1 = bias ? bias[col + 16] : 0.f;

    const v8f* accs[4][2] = { {&acc00, &acc01}, {&acc10, &acc11},
                              {&acc20, &acc21}, {&acc30, &acc31} };
    #pragma unroll
    for (int i = 0; i < 4; ++i) {
        const int rbase = m0 + i * 16 + rhalf;
        #pragma unroll
        for (int r = 0; r < 8; ++r) {
            float x0 = ((*accs[i][0])[r] + bi0) * scale;
            float x1 = ((*accs[i][1])[r] + bi1) * scale;
            if (relu) { x0 = fmaxf(x0, 0.f); x1 = fmaxf(x1, 0.f); }
            const size_t o = (size_t)(rbase + r) * N + col;
            if (addend) { x0 += addend[o]; x1 += addend[o + 16]; }
            out[o] = x0; out[o + 16] = x1;
        }
    }
}

// -----------------------------------------------------------------------------
// In-place softmax over the last dim (=32 = NL*NP): one wave32 per row.
// -----------------------------------------------------------------------------
__global__ __launch_bounds__(256) void softmax32_kernel(float* __restrict__ data, int rows)
{
    const int lane = threadIdx.x & 31;
    const int row  = blockIdx.x * (blockDim.x >> 5) + (threadIdx.x >> 5);
    if (row >= rows) return;
    const size_t base = (size_t)row * 32;
    float x = data[base + lane];
    float m = x;
    #pragma unroll
    for (int off = 16; off > 0; off >>= 1) m = fmaxf(m, __shfl_xor(m, off, 32));
    const float e = __expf(x - m);
    float s = e;
    #pragma unroll
    for (int off = 16; off > 0; off >>= 1) s += __shfl_xor(s, off, 32);
    data[base + lane] = e / s;
}

// -----------------------------------------------------------------------------
// Deformable bilinear sampling + attention-weighted accumulation.
// One wave per (b, q, h). Phase 1: lane = sample index s in [0,32) computes
// coords/weights. Phase 2: loop over samples; broadcast sample t via __shfl,
// all 32 lanes (lane = channel d, HD==32) gather 4 corners coalesced.
//   v:    (B, Lv, NH*HD) row-major (value @ Wv + bv)
//   acc:  (B, LQ, NH, HD) == (B, LQ, C) flat
// -----------------------------------------------------------------------------
__global__ __launch_bounds__(256) void deform_sample_kernel(
    const float* __restrict__ v, const float* __restrict__ refp,
    const float* __restrict__ off, const float* __restrict__ aw,
    const int* __restrict__ shapes, const int* __restrict__ starts,
    float* __restrict__ acc)
{
    const int lane = threadIdx.x & 31;
    const int wave = blockIdx.x * (blockDim.x >> 5) + (threadIdx.x >> 5);
    if (wave >= B_ * LQ_ * NH_) return;             // wave-uniform guard
    const int h  = wave & (NH_ - 1);
    const int lq = (wave / NH_) & (LQ_ - 1);
    const int b  = wave / (NH_ * LQ_);

    // ---- Phase 1: per-lane sample coordinate math (s = lane) ----
    const int s   = lane;
    const int lvl = s >> 3;                         // NP_ == 8
    const int Hl  = shapes[lvl * 2 + 0];
    const int Wl  = shapes[lvl * 2 + 1];
    const int st  = starts[lvl];

    const size_t q  = (size_t)b * LQ_ + lq;
    const float  rx = refp[(q * NL_ + lvl) * 2 + 0];
    const float  ry = refp[(q * NL_ + lvl) * 2 + 1];
    const size_t ob = ((q * NH_ + h) * (NL_ * NP_) + s) * 2;
    const float  ox = off[ob], oy = off[ob + 1];
    const float  wt = aw[(q * NH_ + h) * (NL_ * NP_) + s];

    const float gx  = (rx + ox / (float)Wl) * 2.f - 1.f;
    const float gy  = (ry + oy / (float)Hl) * 2.f - 1.f;
    const float xf  = ((gx + 1.f) * (float)Wl - 1.f) * 0.5f;
    const float yf  = ((gy + 1.f) * (float)Hl - 1.f) * 0.5f;
    const float x0f = floorf(xf), y0f = floorf(yf);
    const float lx  = xf - x0f,   ly  = yf - y0f;
    const int   x0  = (int)x0f,   y0  = (int)y0f;

    // ---- Phase 2: broadcast each sample, gather channel = lane ----
    float a = 0.f;
    const float* vb = v + (size_t)b * LV_ * C_ + h * HD_ + lane;
    #pragma unroll 4
    for (int t = 0; t < 32; ++t) {
        const int   X0 = __shfl(x0, t, 32);
        const int   Y0 = __shfl(y0, t, 32);
        const float LX = __shfl(lx, t, 32);
        const float LY = __shfl(ly, t, 32);
        const float WT = __shfl(wt, t, 32);
        const int   WW = __shfl(Wl, t, 32);
        const int   HH = __shfl(Hl, t, 32);
        const int   ST = __shfl(st, t, 32);
        const int X1 = X0 + 1, Y1 = Y0 + 1;
        const float mx0 = (X0 >= 0 && X0 < WW) ? 1.f : 0.f;
        const float mx1 = (X1 >= 0 && X1 < WW) ? 1.f : 0.f;
        const float my0 = (Y0 >= 0 && Y0 < HH) ? 1.f : 0.f;
        const float my1 = (Y1 >= 0 && Y1 < HH) ? 1.f : 0.f;
        const int cx0 = min(max(X0, 0), WW - 1), cx1 = min(max(X1, 0), WW - 1);
        const int cy0 = min(max(Y0, 0), HH - 1), cy1 = min(max(Y1, 0), HH - 1);
        const float w00 = WT * (1.f - LX) * (1.f - LY) * mx0 * my0;
        const float w10 = WT * LX         * (1.f - LY) * mx1 * my0;
        const float w01 = WT * (1.f - LX) * LY         * mx0 * my1;
        const float w11 = WT * LX         * LY         * mx1 * my1;
        const float v00 = vb[(size_t)(ST + cy0 * WW + cx0) * C_];
        const float v10 = vb[(size_t)(ST + cy0 * WW + cx1) * C_];
        const float v01 = vb[(size_t)(ST + cy1 * WW + cx0) * C_];
        const float v11 = vb[(size_t)(ST + cy1 * WW + cx1) * C_];
        a += w00 * v00 + w10 * v10 + w01 * v01 + w11 * v11;
    }
    acc[(q * NH_ + h) * HD_ + lane] = a;
}

// -----------------------------------------------------------------------------
extern "C" void kernel_launch(void* const* d_in, const int* in_sizes, int n_in,
                              void* d_out, int out_size, void* d_ws, size_t ws_size,
                              hipStream_t stream)
{
    (void)in_sizes; (void)n_in; (void)out_size; (void)ws_size;
    const float* query  = (const float*)d_in[0];
    const float* refp   = (const float*)d_in[1];
    const float* value  = (const float*)d_in[2];
    const int*   shapes = (const int*)  d_in[3];
    const int*   starts = (const int*)  d_in[4];
    const float* W_off  = (const float*)d_in[5];
    const float* b_off  = (const float*)d_in[6];
    const float* W_attn = (const float*)d_in[7];
    const float* b_attn = (const float*)d_in[8];
    const float* Wa1    = (const float*)d_in[9];
    const float* ba1    = (const float*)d_in[10];
    const float* Wa2    = (const float*)d_in[11];
    const float* ba2    = (const float*)d_in[12];
    const float* Wv     = (const float*)d_in[13];
    const float* bv     = (const float*)d_in[14];
    const float* Wo     = (const float*)d_in[15];
    const float* bo     = (const float*)d_in[16];
    float* out = (float*)d_out;

    // Workspace layout (floats): ~127 MB total
    float* v   = (float*)d_ws;                                   // B*Lv*C
    float* off = v   + (size_t)B_ * LV_ * C_;                    // B*LQ*NH*NL*NP*2
    float* hid = off + (size_t)B_ * LQ_ * NH_ * NL_ * NP_ * 2;   // B*LQ*(C/2)
    float* aw  = hid + (size_t)B_ * LQ_ * (C_ / 2);              // B*LQ*NH*NL*NP
    float* acc = aw  + (size_t)B_ * LQ_ * NH_ * NL_ * NP_;       // B*LQ*C

    auto gemm = [&](const float* Ain, const float* Wm, const float* bi,
                    const float* add, float* o, int M, int N, int K,
                    float sc, int relu) {
        const int tiles  = (M / 64) * (N / 32);
        const int blocks = (tiles + 7) / 8;                      // 8 waves / block
        hipLaunchKernelGGL(gemm_f32_wmma, dim3(blocks), dim3(256), 0, stream,
                           Ain, Wm, bi, add, o, M, N, K, sc, relu);
    };

    const int MQ = B_ * LQ_;

    // v = value @ Wv + bv                       (87040 x 256 x 256)
    gemm(value, Wv, bv, nullptr, v, B_ * LV_, C_, C_, 1.f, 0);
    // off = query @ W_off + b_off               (8192 x 512 x 256)
    gemm(query, W_off, b_off, nullptr, off, MQ, NH_ * NL_ * NP_ * 2, C_, 1.f, 0);
    // hid = relu(query @ Wa1 + ba1)             (8192 x 128 x 256)
    gemm(query, Wa1, ba1, nullptr, hid, MQ, C_ / 2, C_, 1.f, 1);
    // off += 0.1 * (hid @ Wa2 + ba2)            (8192 x 512 x 128)
    gemm(hid, Wa2, ba2, off, off, MQ, NH_ * NL_ * NP_ * 2, C_ / 2, 0.1f, 0);
    // aw_logits = query @ W_attn + b_attn       (8192 x 256 x 256)
    gemm(query, W_attn, b_attn, nullptr, aw, MQ, NH_ * NL_ * NP_, C_, 1.f, 0);

    const int rows = MQ * NH_;                                   // 65536
    softmax32_kernel<<<(rows + 7) / 8, 256, 0, stream>>>(aw, rows);
    deform_sample_kernel<<<(rows + 7) / 8, 256, 0, stream>>>(
        v, refp, off, aw, shapes, starts, acc);

    // out = acc @ Wo + bo                       (8192 x 256 x 256)
    gemm(acc, Wo, bo, nullptr, out, MQ, C_, C_, 1.f, 0);
}